// IFNeuron_32779190403424
// MI455X (gfx1250) — compile-verified
//
#include <hip/hip_runtime.h>

// ST-BIF IF neuron, multistep scan over T=16.
// x_seq (T=16, B=64, N=196, D=768) f32 -> out, same shape/dtype.
// Pure HBM-streaming kernel (~1.23 GB moved exactly once => ~53 us roofline
// at 23.3 TB/s). Each thread owns 4 contiguous (B,N,D) elements and walks all
// 16 timesteps with the (q, acc) recurrence held entirely in VGPRs.
//
// CDNA5 specifics: wave32 codegen, global_load/store_b128 with th:NT
// (single-use stream, keep the 192MB L2 clean), split s_wait_* counters, and
// 16 distinct SGPR base pairs (pinned by one multi-operand "+s" asm barrier)
// + one divergent scaled 32-bit voffset so the 16-deep load burst issues as
// back-to-back global_load_b128 saddr+voffset with no v_add_u64 chains and no
// s_wait_xcnt serialization. addrspace(1) casts keep the ops GLOBAL (not
// FLAT, which would also tie up DScnt).

typedef __attribute__((ext_vector_type(4))) float v4f;
typedef const v4f __attribute__((address_space(1)))* gv4f_cptr;
typedef v4f __attribute__((address_space(1)))* gv4f_ptr;

#define TSTEPS 16
#define POS_MAX 7.0f   // LEVEL/2 - 1
#define NEG_MIN -8.0f  // -(LEVEL/2)

__device__ __forceinline__ float stbif_step(float xin, float inv_qt, float qt,
                                            float& q, float& acc) {
  q += xin * inv_qt;
  // acc is always an exact small integer (starts at 0, changes by +-1), so
  // the reference's round(acc) is the identity and is elided.
  const bool pos = (q - 1.0f >= 0.0f) && (acc < POS_MAX);
  const bool neg = (q < 0.0f) && (acc > NEG_MIN);
  const float cur = pos ? 1.0f : (neg ? -1.0f : 0.0f);
  acc += cur;
  q -= cur;
  return cur * qt;
}

__global__ __launch_bounds__(256) void stbif_vec4(
    const char* __restrict__ xb, const float* __restrict__ qt_ptr,
    char* __restrict__ ob, long long n_vec, long long stride_bytes) {
  const long long li = (long long)blockIdx.x * blockDim.x + threadIdx.x;
  if (li >= n_vec) return;

  // Divergent byte offset within one timestep plane (38.5 MB for the
  // reference shape => provably fits 32 bits => eligible for GVS voffset).
  const unsigned voff = (unsigned)(li * 16ll);

  const float qt = qt_ptr[0];
  const float inv_qt = 1.0f / qt;  // exact for qt = 0.5

  // All 16 per-timestep input bases, pinned into 16 *distinct* SGPR pairs by
  // a single asm barrier (simultaneous live ranges). This both blocks the
  // (base + t*stride) + voff -> (base + voff) + t*stride reassociation and
  // guarantees no saddr register is overwritten while its load is in flight.
  unsigned long long xbase[TSTEPS];
#pragma unroll
  for (int t = 0; t < TSTEPS; ++t)
    xbase[t] = (unsigned long long)xb +
               (unsigned long long)t * (unsigned long long)stride_bytes;
  asm("" : "+s"(xbase[0]), "+s"(xbase[1]), "+s"(xbase[2]), "+s"(xbase[3]),
            "+s"(xbase[4]), "+s"(xbase[5]), "+s"(xbase[6]), "+s"(xbase[7]),
            "+s"(xbase[8]), "+s"(xbase[9]), "+s"(xbase[10]), "+s"(xbase[11]),
            "+s"(xbase[12]), "+s"(xbase[13]), "+s"(xbase[14]), "+s"(xbase[15]));

  // 16 outstanding global_load_b128 per wave (max MLP), each saddr+voffset.
  v4f v[TSTEPS];
#pragma unroll
  for (int t = 0; t < TSTEPS; ++t)
    v[t] = __builtin_nontemporal_load((gv4f_cptr)(xbase[t] + voff));

  // Same treatment for the output bases.
  unsigned long long obase[TSTEPS];
#pragma unroll
  for (int t = 0; t < TSTEPS; ++t)
    obase[t] = (unsigned long long)ob +
               (unsigned long long)t * (unsigned long long)stride_bytes;
  asm("" : "+s"(obase[0]), "+s"(obase[1]), "+s"(obase[2]), "+s"(obase[3]),
            "+s"(obase[4]), "+s"(obase[5]), "+s"(obase[6]), "+s"(obase[7]),
            "+s"(obase[8]), "+s"(obase[9]), "+s"(obase[10]), "+s"(obase[11]),
            "+s"(obase[12]), "+s"(obase[13]), "+s"(obase[14]), "+s"(obase[15]));

  float q0 = 0.5f, q1 = 0.5f, q2 = 0.5f, q3 = 0.5f;
  float a0 = 0.0f, a1 = 0.0f, a2 = 0.0f, a3 = 0.0f;

#pragma unroll
  for (int t = 0; t < TSTEPS; ++t) {
    v4f r;
    r.x = stbif_step(v[t].x, inv_qt, qt, q0, a0);
    r.y = stbif_step(v[t].y, inv_qt, qt, q1, a1);
    r.z = stbif_step(v[t].z, inv_qt, qt, q2, a2);
    r.w = stbif_step(v[t].w, inv_qt, qt, q3, a3);
    __builtin_nontemporal_store(r, (gv4f_ptr)(obase[t] + voff));
  }
}

// Scalar tail for per-timestep element counts not divisible by 4 (not hit for
// the reference shape, but keeps the kernel shape-generic).
__global__ __launch_bounds__(256) void stbif_scalar(
    const float* __restrict__ x, const float* __restrict__ qt_ptr,
    float* __restrict__ out, long long n, long long stride) {
  const long long i = (long long)blockIdx.x * blockDim.x + threadIdx.x;
  if (i >= n) return;

  const float qt = qt_ptr[0];
  const float inv_qt = 1.0f / qt;

  float q = 0.5f, a = 0.0f;
#pragma unroll
  for (int t = 0; t < TSTEPS; ++t) {
    const float xin = __builtin_nontemporal_load(x + (long long)t * stride + i);
    __builtin_nontemporal_store(stbif_step(xin, inv_qt, qt, q, a),
                                out + (long long)t * stride + i);
  }
}

extern "C" void kernel_launch(void* const* d_in, const int* in_sizes, int n_in,
                              void* d_out, int out_size, void* d_ws, size_t ws_size,
                              hipStream_t stream) {
  const float* x = (const float*)d_in[0];        // (T,B,N,D) f32
  const float* qt = (const float*)d_in[1];       // scalar q_threshold
  float* out = (float*)d_out;

  const long long total = (long long)in_sizes[0];
  const long long M = total / TSTEPS;            // elements per timestep
  const long long n_vec = M / 4;                 // float4 columns
  const long long rem = M - n_vec * 4;

  if (n_vec > 0) {
    const int block = 256;
    const long long grid = (n_vec + block - 1) / block;
    stbif_vec4<<<(unsigned int)grid, block, 0, stream>>>(
        (const char*)x, qt, (char*)out, n_vec, M * 4ll);
  }
  if (rem > 0) {
    stbif_scalar<<<1, 256, 0, stream>>>(x + n_vec * 4, qt, out + n_vec * 4,
                                        rem, M);
  }
}